// JoinSegmenter_36515811951134
// MI455X (gfx1250) — compile-verified
//
#include <hip/hip_runtime.h>
#include <hip/hip_bf16.h>

// JoinSegmenter for MI455X (gfx1250, wave32).
// B=32, T=4096, C=256. Memory-bound (~0.4 GB traffic -> ~17us @ 23.3 TB/s).
// Segment-mean runs on the matrix core as D(16x16) += A(16x4)*L(4x16) with
// V_WMMA_F32_16X16X4_F32 (A = selection/averaging weights, exact f32).
// Loads in the WMMA loop are address-clamped (never predicated) and
// software-pipelined one chunk ahead so the loop is branch-free.

#define BB 32
#define TT 4096
#define CC 256

typedef __attribute__((ext_vector_type(2))) float v2f;
typedef __attribute__((ext_vector_type(8))) float v8f;

// ---------------------------------------------------------------- kernel 1
// One wave per (b,t) row: lane handles 8 contiguous channels (coalesced 1KB),
// first-max tie-break matches jnp.argmax.
__global__ void js_argmax_kernel(const float* __restrict__ logits,
                                 const unsigned char* __restrict__ mask,
                                 int* __restrict__ preds) {
    int waveInBlk = threadIdx.x >> 5;
    int lane      = threadIdx.x & 31;
    int row       = blockIdx.x * (blockDim.x >> 5) + waveInBlk;   // b*T + t
    const float* p = logits + (size_t)row * CC;

    int cbase = lane * 8;
    float4 a = *(const float4*)(p + cbase);
    float4 b = *(const float4*)(p + cbase + 4);
    float vals[8] = {a.x, a.y, a.z, a.w, b.x, b.y, b.z, b.w};
    float bv = vals[0];
    int   bi = cbase;
#pragma unroll
    for (int i = 1; i < 8; ++i)
        if (vals[i] > bv) { bv = vals[i]; bi = cbase + i; }

#pragma unroll
    for (int m = 16; m >= 1; m >>= 1) {
        float ov = __shfl_xor(bv, m, 32);
        int   oi = __shfl_xor(bi, m, 32);
        if (ov > bv || (ov == bv && oi < bi)) { bv = ov; bi = oi; }
    }
    if (lane == 0)
        preds[row] = mask[row] ? -1 : bi;
}

// ---------------------------------------------------------------- kernel 2
// One 256-thread block per batch. Two LDS block-scans:
//  (a) boundary scan -> run_start / run_val,  (b) keep-compaction -> kept_map.
__device__ __forceinline__ int js_block_scan(int val, int* sh, int tid) {
    sh[tid] = val;
    __syncthreads();
    for (int off = 1; off < 256; off <<= 1) {
        int v = (tid >= off) ? sh[tid - off] : 0;
        __syncthreads();
        sh[tid] += v;
        __syncthreads();
    }
    return sh[tid];                     // inclusive scan; sh[255] = total
}

__global__ void js_segment_kernel(const int* __restrict__ preds,
                                  int* __restrict__ run_start,
                                  int* __restrict__ run_len,
                                  int* __restrict__ run_val,
                                  int* __restrict__ kept_map,
                                  int* __restrict__ n_keep) {
    __shared__ int sh[256];
    int b   = blockIdx.x;
    int tid = threadIdx.x;
    int t0  = tid * 16;
    const int* p = preds + (size_t)b * TT;

    // ---- phase A: run boundaries
    int myPred[16];
#pragma unroll
    for (int k = 0; k < 16; ++k) myPred[k] = p[t0 + k];
    int prev = (t0 == 0) ? (int)0x80000000 : p[t0 - 1];
    unsigned bmask = 0;
    int cnt = 0;
#pragma unroll
    for (int k = 0; k < 16; ++k) {
        int pk = (k == 0) ? prev : myPred[k - 1];
        if (myPred[k] != pk) { bmask |= 1u << k; ++cnt; }
    }
    int incl  = js_block_scan(cnt, sh, tid);
    int nruns = sh[255];
    int seg   = incl - cnt;
#pragma unroll
    for (int k = 0; k < 16; ++k) {
        if (bmask & (1u << k)) {
            run_start[(size_t)b * TT + seg] = t0 + k;
            run_val  [(size_t)b * TT + seg] = myPred[k];
            ++seg;
        }
    }
    __syncthreads();   // run_start/run_val visible; LDS reusable

    // ---- phase B: lengths + keep compaction
    unsigned kmask = 0;
    int cnt2 = 0;
#pragma unroll
    for (int k = 0; k < 16; ++k) {
        int s = t0 + k;
        if (s < nruns) {
            int st  = run_start[(size_t)b * TT + s];
            int nxt = (s + 1 < nruns) ? run_start[(size_t)b * TT + s + 1] : TT;
            run_len[(size_t)b * TT + s] = nxt - st;
            if (run_val[(size_t)b * TT + s] != -1) { kmask |= 1u << k; ++cnt2; }
        }
    }
    int incl2 = js_block_scan(cnt2, sh, tid);
    int nk    = sh[255];
    int d     = incl2 - cnt2;
#pragma unroll
    for (int k = 0; k < 16; ++k) {
        if (kmask & (1u << k)) {
            kept_map[(size_t)b * TT + d] = t0 + k;
            ++d;
        }
    }
    if (tid == 0) n_keep[b] = nk;
}

// ---------------------------------------------------------------- kernel 3
// Wave = (batch, 16-output-row tile, 16-channel tile). Accumulate
// D(16x16) += A(16x4) * L(4x16) over the tile's frame span with f32 WMMA.
// Out-of-range frames are neutralized by zero A entries, so B loads are
// merely address-clamped (branch-free) and pipelined one chunk ahead.
__global__ void js_means_kernel(const float* __restrict__ logits,
                                const int* __restrict__ run_start,
                                const int* __restrict__ run_len,
                                const int* __restrict__ kept_map,
                                const int* __restrict__ n_keep,
                                float* __restrict__ out) {
    int b    = blockIdx.z;
    int tile = blockIdx.y;                       // 16-row output tile
    int ct   = blockIdx.x * 4 + threadIdx.y;     // 16-channel tile (0..15)
    int lane = threadIdx.x;                      // wave32 lane
    int m    = lane & 15;

    int nk = n_keep[b];
    int j0 = tile * 16;
    v8f acc = {};

    if (j0 < nk) {
        int  j        = j0 + m;
        bool rowValid = j < nk;
        int  s  = rowValid ? kept_map[(size_t)b * TT + j] : 0;
        int  st = rowValid ? run_start[(size_t)b * TT + s] : 0;
        int  ln = rowValid ? run_len  [(size_t)b * TT + s] : 0;
        int  en = rowValid ? st + ln : 0;
        float w = rowValid ? 1.0f / (float)ln : 0.0f;

        int spanStart = __shfl(st, 0, 32);       // row j0 is valid (j0 < nk)
        int spanEnd   = en;
#pragma unroll
        for (int mk = 16; mk >= 1; mk >>= 1)
            spanEnd = max(spanEnd, __shfl_xor(spanEnd, mk, 32));

        const float* Lb = logits + (size_t)b * TT * CC + (ct * 16 + m);
        int kA = (lane < 16) ? 0 : 2;            // A/B VGPR0<->K{0,2}, VGPR1<->K{1,3}

        // pipeline prologue: chunk at spanStart
        int   ca0 = min(spanStart + kA,     TT - 1);
        int   ca1 = min(spanStart + kA + 1, TT - 1);
        float cb0 = Lb[(size_t)ca0 * CC];
        float cb1 = Lb[(size_t)ca1 * CC];

        for (int t0 = spanStart; t0 < spanEnd; t0 += 4) {
            // issue next chunk's loads (clamped; harmless overrun by one chunk)
            int   na0 = min(t0 + 4 + kA,     TT - 1);
            int   na1 = min(t0 + 4 + kA + 1, TT - 1);
            float nb0 = Lb[(size_t)na0 * CC];
            float nb1 = Lb[(size_t)na1 * CC];

            int ta = t0 + kA;
            int tb = ta + 1;
            v2f A, Bv;
            A.x  = (ta >= st && ta < en) ? w : 0.0f;
            A.y  = (tb >= st && tb < en) ? w : 0.0f;
            Bv.x = cb0;
            Bv.y = cb1;
            acc = __builtin_amdgcn_wmma_f32_16x16x4_f32(
                      false, A, false, Bv, (short)0, acc, false, false);

            cb0 = nb0;
            cb1 = nb1;
        }
    }

    // D layout: VGPR r -> rows r (lanes 0-15) and r+8 (lanes 16-31), N = lane&15
    int rowOff = (lane < 16) ? 0 : 8;
#pragma unroll
    for (int r = 0; r < 8; ++r) {
        int orow = j0 + r + rowOff;
        out[((size_t)b * TT + orow) * CC + ct * 16 + m] = acc[r];
    }
}

// ---------------------------------------------------------------- kernel 4
__global__ void js_pad_kernel(const int* __restrict__ n_keep,
                              float* __restrict__ padOut) {
    int idx = blockIdx.x * blockDim.x + threadIdx.x;   // b*T + t
    int b = idx >> 12;
    int t = idx & (TT - 1);
    padOut[idx] = (t >= n_keep[b]) ? 1.0f : 0.0f;
}

// ---------------------------------------------------------------- launch
extern "C" void kernel_launch(void* const* d_in, const int* in_sizes, int n_in,
                              void* d_out, int out_size, void* d_ws, size_t ws_size,
                              hipStream_t stream) {
    const float*         logits = (const float*)d_in[0];
    const unsigned char* mask   = (const unsigned char*)d_in[1];  // jnp.bool_ = 1 byte
    float* out = (float*)d_out;

    int* preds     = (int*)d_ws;                 // [B*T]
    int* run_start = preds     + BB * TT;        // [B*T]
    int* run_len   = run_start + BB * TT;        // [B*T]
    int* run_val   = run_len   + BB * TT;        // [B*T]
    int* kept_map  = run_val   + BB * TT;        // [B*T]
    int* n_keep    = kept_map  + BB * TT;        // [B]      total ~2.5 MB

    // 1) argmax: 8 waves/block, one wave per (b,t)
    js_argmax_kernel<<<(BB * TT) / 8, 256, 0, stream>>>(logits, mask, preds);
    // 2) RLE + compaction: one block per batch
    js_segment_kernel<<<BB, 256, 0, stream>>>(preds, run_start, run_len,
                                              run_val, kept_map, n_keep);
    // 3) WMMA segment means: grid (4 chan-tile groups, 256 row tiles, 32 batches)
    js_means_kernel<<<dim3(4, TT / 16, BB), dim3(32, 4, 1), 0, stream>>>(
        logits, run_start, run_len, kept_map, n_keep, out);
    // 4) pad flags (as f32, appended after B*T*C logits)
    js_pad_kernel<<<(BB * TT) / 256, 256, 0, stream>>>(
        n_keep, out + (size_t)BB * TT * CC);
}